// GAT_VGAE_47433618817225
// MI455X (gfx1250) — compile-verified
//
#include <hip/hip_runtime.h>
#include <hip/hip_bf16.h>

typedef __attribute__((ext_vector_type(2))) float v2f;
typedef __attribute__((ext_vector_type(8))) float v8f;

#define NN    1024
#define FIN   384
#define HC    1024      // H*C1 = 8*128
#define HEADS 8
#define C1    128
#define EMB   64
#define EDGES 32768
#define ETOT  (EDGES + NN)   // with self-loops
#define OUTSZ (FIN * FIN)    // 147456

__device__ __forceinline__ float lrelu(float v) { return v > 0.f ? v : 0.2f * v; }

__device__ __forceinline__ void atomicMaxF(float* addr, float val) {
  unsigned int* ua = (unsigned int*)addr;
  unsigned int old = *ua;
  for (;;) {
    float f = __uint_as_float(old);
    if (f >= val) break;
    unsigned int assumed = old;
    old = atomicCAS(ua, assumed, __float_as_uint(val));
    if (old == assumed) break;
  }
}

// ---------------------------------------------------------------- init
__global__ void init_ws_kernel(float* h0acc, float* emax, float* denom, int* needed) {
  int i = blockIdx.x * blockDim.x + threadIdx.x;
  if (i < NN * HC) h0acc[i] = 0.f;
  if (i < NN * HEADS) { emax[i] = -__builtin_inff(); denom[i] = 0.f; }
  if (i < NN) needed[i] = 0;
}

// ------------------------------------------------- mark nodes feeding mu[0]
__global__ void mark_needed_kernel(const int* __restrict__ ei, int* __restrict__ needed) {
  int e = blockIdx.x * blockDim.x + threadIdx.x;
  if (e == 0) needed[0] = 1;                 // self-loop (0,0)
  if (e < EDGES && ei[EDGES + e] == 0) needed[ei[e]] = 1;
}

// ----------------------------------- GEMM1: xw1[n,m] = sum_k xf[k,n]*W1[k,m]
// fp32 WMMA 16x16x4; each wave computes a 16x64 output strip (A-frag reuse x4).
__global__ __launch_bounds__(256) void gemm1_wmma_kernel(const float* __restrict__ xf,
                                                         const float* __restrict__ W1,
                                                         float* __restrict__ xw1) {
  const int wave = (blockIdx.x * blockDim.x + threadIdx.x) >> 5;  // 1024 waves
  const int lane = threadIdx.x & 31;
  const int tileM  = wave >> 4;       // 0..63
  const int tileN4 = wave & 15;       // 0..15
  const int m0 = tileM * 16;
  const int n0 = tileN4 * 64;
  const int mrow = m0 + (lane & 15);
  const int khalf = (lane >> 4) << 1; // 0 for lanes 0-15, 2 for lanes 16-31
  v8f acc0 = {}, acc1 = {}, acc2 = {}, acc3 = {};
  for (int k = 0; k < FIN; k += 4) {
    const int ka = k + khalf;
    v2f a;
    a.x = xf[(ka + 0) * NN + mrow];   // x = x_features.T : A[m,k] = xf[k*N + m]
    a.y = xf[(ka + 1) * NN + mrow];
    const int ncol = n0 + (lane & 15);
    v2f b0, b1, b2, b3;
    b0.x = W1[(ka + 0) * HC + ncol +  0]; b0.y = W1[(ka + 1) * HC + ncol +  0];
    b1.x = W1[(ka + 0) * HC + ncol + 16]; b1.y = W1[(ka + 1) * HC + ncol + 16];
    b2.x = W1[(ka + 0) * HC + ncol + 32]; b2.y = W1[(ka + 1) * HC + ncol + 32];
    b3.x = W1[(ka + 0) * HC + ncol + 48]; b3.y = W1[(ka + 1) * HC + ncol + 48];
    acc0 = __builtin_amdgcn_wmma_f32_16x16x4_f32(false, a, false, b0, (short)0, acc0, false, false);
    acc1 = __builtin_amdgcn_wmma_f32_16x16x4_f32(false, a, false, b1, (short)0, acc1, false, false);
    acc2 = __builtin_amdgcn_wmma_f32_16x16x4_f32(false, a, false, b2, (short)0, acc2, false, false);
    acc3 = __builtin_amdgcn_wmma_f32_16x16x4_f32(false, a, false, b3, (short)0, acc3, false, false);
  }
  const int rbase = m0 + ((lane >> 4) << 3);   // lanes 0-15 -> M=v, 16-31 -> M=v+8
  const int cbase = n0 + (lane & 15);
#pragma unroll
  for (int v = 0; v < 8; ++v) {
    xw1[(rbase + v) * HC + cbase +  0] = acc0[v];
    xw1[(rbase + v) * HC + cbase + 16] = acc1[v];
    xw1[(rbase + v) * HC + cbase + 32] = acc2[v];
    xw1[(rbase + v) * HC + cbase + 48] = acc3[v];
  }
}

// ---------------------------------------- layer-1 attention coefficients
__global__ __launch_bounds__(256) void attn1_kernel(const float* __restrict__ xw1,
                                                    const float* __restrict__ a_src1,
                                                    const float* __restrict__ a_dst1,
                                                    float* __restrict__ as1, float* __restrict__ ad1) {
  const int n = blockIdx.x;
  const int h = threadIdx.x >> 5;
  const int lane = threadIdx.x & 31;
  float s = 0.f, d = 0.f;
  for (int c = lane; c < C1; c += 32) {
    float v = xw1[n * HC + h * C1 + c];
    s += v * a_src1[h * C1 + c];
    d += v * a_dst1[h * C1 + c];
  }
  for (int off = 16; off; off >>= 1) { s += __shfl_down(s, off); d += __shfl_down(d, off); }
  if (lane == 0) { as1[n * HEADS + h] = s; ad1[n * HEADS + h] = d; }
}

// --------------------------------------------- layer-1 edge passes (pruned)
__global__ void edge_max1_kernel(const int* __restrict__ ei, const int* __restrict__ needed,
                                 const float* __restrict__ as1, const float* __restrict__ ad1,
                                 float* __restrict__ emax) {
  int e = blockIdx.x * blockDim.x + threadIdx.x;
  if (e >= ETOT) return;
  int src = e < EDGES ? ei[e] : (e - EDGES);
  int dst = e < EDGES ? ei[EDGES + e] : (e - EDGES);
  if (!needed[dst]) return;
#pragma unroll
  for (int h = 0; h < HEADS; ++h)
    atomicMaxF(&emax[dst * HEADS + h], lrelu(as1[src * HEADS + h] + ad1[dst * HEADS + h]));
}

__global__ void edge_denom1_kernel(const int* __restrict__ ei, const int* __restrict__ needed,
                                   const float* __restrict__ as1, const float* __restrict__ ad1,
                                   const float* __restrict__ emax, float* __restrict__ denom) {
  int e = blockIdx.x * blockDim.x + threadIdx.x;
  if (e >= ETOT) return;
  int src = e < EDGES ? ei[e] : (e - EDGES);
  int dst = e < EDGES ? ei[EDGES + e] : (e - EDGES);
  if (!needed[dst]) return;
#pragma unroll
  for (int h = 0; h < HEADS; ++h) {
    float v = lrelu(as1[src * HEADS + h] + ad1[dst * HEADS + h]);
    atomicAdd(&denom[dst * HEADS + h], __expf(v - emax[dst * HEADS + h]));
  }
}

__global__ __launch_bounds__(256) void edge_scatter1_kernel(const int* __restrict__ ei,
    const int* __restrict__ needed, const float* __restrict__ as1, const float* __restrict__ ad1,
    const float* __restrict__ emax, const float* __restrict__ denom,
    const float* __restrict__ xw1, float* __restrict__ h0acc) {
  int e = blockIdx.x;
  int src = e < EDGES ? ei[e] : (e - EDGES);
  int dst = e < EDGES ? ei[EDGES + e] : (e - EDGES);
  if (!needed[dst]) return;
  __shared__ float alpha[HEADS];
  if (threadIdx.x < HEADS) {
    int h = threadIdx.x;
    float v = lrelu(as1[src * HEADS + h] + ad1[dst * HEADS + h]);
    alpha[h] = __expf(v - emax[dst * HEADS + h]) / (denom[dst * HEADS + h] + 1e-16f);
  }
  __syncthreads();
  for (int c = threadIdx.x; c < HC; c += 256)
    atomicAdd(&h0acc[dst * HC + c], xw1[src * HC + c] * alpha[c >> 7]);
}

__global__ __launch_bounds__(256) void h0_final_kernel(const int* __restrict__ needed,
                                                       const float* __restrict__ b1,
                                                       float* __restrict__ h0) {
  int n = blockIdx.x;
  if (!needed[n]) return;
  for (int c = threadIdx.x; c < HC; c += 256) {
    float v = h0[n * HC + c] + b1[c];
    h0[n * HC + c] = v > 0.f ? v : 0.f;
  }
}

// ---------------- xw2 rows + layer-2 attention coefs (needed nodes only)
__global__ __launch_bounds__(64) void xw2_rows_kernel(const int* __restrict__ needed,
    const float* __restrict__ h0, const float* __restrict__ W2,
    const float* __restrict__ a_src2, const float* __restrict__ a_dst2,
    float* __restrict__ xw2, float* __restrict__ as2, float* __restrict__ ad2) {
  int n = blockIdx.x;
  if (!needed[n]) return;
  int c = threadIdx.x;
  float acc = 0.f;
  for (int k = 0; k < HC; ++k) acc += h0[n * HC + k] * W2[k * EMB + c];
  xw2[n * EMB + c] = acc;
  __shared__ float sb[EMB], db[EMB];
  sb[c] = acc * a_src2[c];
  db[c] = acc * a_dst2[c];
  __syncthreads();
  for (int off = 32; off; off >>= 1) {
    if (c < off) { sb[c] += sb[c + off]; db[c] += db[c + off]; }
    __syncthreads();
  }
  if (c == 0) { as2[n] = sb[0]; ad2[n] = db[0]; }
}

// ---------- node-0 segment softmax + aggregation + mu/z fusion (1 block)
__global__ __launch_bounds__(256) void node0_kernel(const int* __restrict__ ei,
    const float* __restrict__ as2, const float* __restrict__ ad2,
    const float* __restrict__ xw2, const float* __restrict__ b2,
    const float* __restrict__ eps, float* __restrict__ z0) {
  __shared__ float red[256];
  __shared__ float s_max, s_den;
  const int tid = threadIdx.x;
  const float ad0 = ad2[0];
  const float eself = lrelu(as2[0] + ad0);
  // pass 1: segment max over edges with dst==0 (+ self-loop)
  float m = eself;
  for (int e = tid; e < EDGES; e += 256)
    if (ei[EDGES + e] == 0) m = fmaxf(m, lrelu(as2[ei[e]] + ad0));
  red[tid] = m; __syncthreads();
  for (int off = 128; off; off >>= 1) {
    if (tid < off) red[tid] = fmaxf(red[tid], red[tid + off]);
    __syncthreads();
  }
  if (tid == 0) s_max = red[0];
  __syncthreads();
  const float mx = s_max;
  // pass 2: exp-sum
  float s = (tid == 0) ? __expf(eself - mx) : 0.f;
  for (int e = tid; e < EDGES; e += 256)
    if (ei[EDGES + e] == 0) s += __expf(lrelu(as2[ei[e]] + ad0) - mx);
  red[tid] = s; __syncthreads();
  for (int off = 128; off; off >>= 1) {
    if (tid < off) red[tid] += red[tid + off];
    __syncthreads();
  }
  if (tid == 0) s_den = red[0] + 1e-16f;
  __syncthreads();
  const float den = s_den;
  // pass 3: weighted aggregation, 4 groups x 64 channels
  const int c = tid & 63, g = tid >> 6;
  float acc = (g == 0) ? (__expf(eself - mx) / den) * xw2[c] : 0.f;
  for (int e = g; e < EDGES; e += 4)
    if (ei[EDGES + e] == 0) {
      int src = ei[e];
      acc += (__expf(lrelu(as2[src] + ad0) - mx) / den) * xw2[src * EMB + c];
    }
  red[tid] = acc; __syncthreads();
  if (tid < EMB) {
    float t = red[tid] + red[tid + 64] + red[tid + 128] + red[tid + 192];
    float mu = t + b2[tid];
    mu = mu > 0.f ? mu : 0.f;                       // relu
    z0[tid] = mu + eps[tid] * __expf(mu);           // z = mu + eps*exp(mu), row 0
  }
}

// --------------------- decoder GEMV: out = sigmoid(Wd @ z0 + bd), 1 row/wave
__global__ __launch_bounds__(256) void decoder_kernel(const float* __restrict__ Wd,
                                                      const float* __restrict__ bd,
                                                      const float* __restrict__ z0,
                                                      float* __restrict__ out) {
  __shared__ float zs[EMB];
  if (threadIdx.x < EMB) zs[threadIdx.x] = z0[threadIdx.x];
  __syncthreads();
  const int row = blockIdx.x * 8 + (threadIdx.x >> 5);
  const int lane = threadIdx.x & 31;
  const float2 w = ((const float2*)(Wd + (size_t)row * EMB))[lane];
  float p = w.x * zs[lane * 2] + w.y * zs[lane * 2 + 1];
  for (int off = 16; off; off >>= 1) p += __shfl_down(p, off);
  if (lane == 0) out[row] = 1.f / (1.f + __expf(-(p + bd[row])));
}

extern "C" void kernel_launch(void* const* d_in, const int* in_sizes, int n_in,
                              void* d_out, int out_size, void* d_ws, size_t ws_size,
                              hipStream_t stream) {
  const int*   ei   = (const int*)  d_in[0];
  const float* xf   = (const float*)d_in[1];
  const float* W1   = (const float*)d_in[2];
  const float* as1w = (const float*)d_in[3];
  const float* ad1w = (const float*)d_in[4];
  const float* b1   = (const float*)d_in[5];
  const float* W2   = (const float*)d_in[6];
  const float* aS2  = (const float*)d_in[7];
  const float* aD2  = (const float*)d_in[8];
  const float* b2   = (const float*)d_in[9];
  const float* Wd   = (const float*)d_in[10];
  const float* bd   = (const float*)d_in[11];
  const float* eps  = (const float*)d_in[12];
  float* out = (float*)d_out;

  float* w = (float*)d_ws;
  float* xw1   = w;                               // 1024*1024
  float* h0    = xw1 + (size_t)NN * HC;           // 1024*1024
  float* as1   = h0  + (size_t)NN * HC;           // 8192
  float* ad1   = as1 + NN * HEADS;                // 8192
  float* emax  = ad1 + NN * HEADS;                // 8192
  float* denom = emax + NN * HEADS;               // 8192
  float* xw2   = denom + NN * HEADS;              // 65536
  float* as2   = xw2 + NN * EMB;                  // 1024
  float* ad2   = as2 + NN;                        // 1024
  float* z0    = ad2 + NN;                        // 64
  int*   needed = (int*)(z0 + EMB);               // 1024 ints

  init_ws_kernel<<<(NN * HC + 255) / 256, 256, 0, stream>>>(h0, emax, denom, needed);
  mark_needed_kernel<<<(EDGES + 255) / 256, 256, 0, stream>>>(ei, needed);
  gemm1_wmma_kernel<<<128, 256, 0, stream>>>(xf, W1, xw1);
  attn1_kernel<<<NN, 256, 0, stream>>>(xw1, as1w, ad1w, as1, ad1);
  edge_max1_kernel<<<(ETOT + 255) / 256, 256, 0, stream>>>(ei, needed, as1, ad1, emax);
  edge_denom1_kernel<<<(ETOT + 255) / 256, 256, 0, stream>>>(ei, needed, as1, ad1, emax, denom);
  edge_scatter1_kernel<<<ETOT, 256, 0, stream>>>(ei, needed, as1, ad1, emax, denom, xw1, h0);
  h0_final_kernel<<<NN, 256, 0, stream>>>(needed, b1, h0);
  xw2_rows_kernel<<<NN, 64, 0, stream>>>(needed, h0, W2, aS2, aD2, xw2, as2, ad2);
  node0_kernel<<<1, 256, 0, stream>>>(ei, as2, ad2, xw2, b2, eps, z0);
  decoder_kernel<<<OUTSZ / 8, 256, 0, stream>>>(Wd, bd, z0, out);
}